// Voroloss_opt_81286551044464
// MI455X (gfx1250) — compile-verified
//
#include <hip/hip_runtime.h>
#include <math.h>

typedef __attribute__((ext_vector_type(2))) float v2f;
typedef __attribute__((ext_vector_type(8))) float v8f;

#define NPTS 8192
#define KNN  16
#define NTILE (NPTS / 16)
#define PTS_PER_BLOCK 64   // 4 waves * 16 points
#define BLOCK_THREADS 128

__global__ void voroloss_kernel(const float* __restrict__ points,
                                const float* __restrict__ spoints,
                                float* __restrict__ out)
{
    // LDS: per spoint m, float4 = (-2x, -2y, -2z, s2)  -> 8192*16 = 128 KB
    extern __shared__ float sraw[];
    float4* sB = (float4*)sraw;

    const int b    = blockIdx.y;
    const int tid  = threadIdx.x;
    const int lane = tid & 31;
    const int wave = tid >> 5;
    const int laneM = lane & 15;
    const int hi    = lane >> 4;      // 0: lanes 0-15, 1: lanes 16-31

    const float* P = points  + (size_t)b * NPTS * 3;
    const float* S = spoints + (size_t)b * NPTS * 3;

    // ---- Stage B-layout spoints into LDS (one ds_store_b128 per spoint) ----
    for (int m = tid; m < NPTS; m += BLOCK_THREADS) {
        float sx = S[3*m+0], sy = S[3*m+1], sz = S[3*m+2];
        float4 v;
        v.x = -2.0f * sx;
        v.y = -2.0f * sy;
        v.z = -2.0f * sz;
        v.w = sx*sx + sy*sy + sz*sz;
        sB[m] = v;
    }
    __syncthreads();

    // ---- Load this wave's 16-point A tile ----
    const int row = blockIdx.x * PTS_PER_BLOCK + wave * 16 + laneM;
    const float px = P[3*row+0];
    const float py = P[3*row+1];
    const float pz = P[3*row+2];

    // 32-bit A-matrix 16x4 layout: VGPR0 = K=0 (lo) / K=2 (hi); VGPR1 = K=1 (lo) / K=3 (hi)
    // A row = (px, py, pz, 1.0), B col = (-2sx, -2sy, -2sz, s2)  =>
    //   C[m][n] = s2[n] - 2*dot(p[m], s[n])   (ranking score; p2 is row-constant)
    v2f a;
    a.x = hi ? pz   : px;
    a.y = hi ? 1.0f : py;

    // ---- Per-lane sorted top-16 (ascending; [15] is current max) ----
    float dl[KNN];
    int   il[KNN];
#pragma unroll
    for (int j = 0; j < KNN; ++j) { dl[j] = 3.0e38f; il[j] = 0; }

    // Software pipeline: B vector for tile t is loaded during tile t-1's
    // candidate processing, so the DS wait is off the WMMA critical path.
    const int lanebase = laneM * 4 + 2 * hi;   // float offset within tile row
    v2f bv;
    {
        const float* src = &sraw[lanebase];
        bv.x = src[0];
        bv.y = src[1];
    }

    for (int t = 0; t < NTILE; ++t) {
        // Prefetch next tile's B vector (wraps to 0 on last iter; harmless).
        const int tn = (t + 1) & (NTILE - 1);
        const float* srcn = &sraw[tn * 64 + lanebase];
        v2f bnext;
        bnext.x = srcn[0];
        bnext.y = srcn[1];

        v8f c = {};
        c = __builtin_amdgcn_wmma_f32_16x16x4_f32(
                /*neg_a=*/false, a, /*neg_b=*/false, bv,
                /*c_mod=*/(short)0, c, /*reuse_a=*/false, /*reuse_b=*/false);

        const int cbase = t * 16 + 8 * hi;   // lane's candidate column base in spoints
#pragma unroll
        for (int j = 0; j < 8; ++j) {
            float v = c[j];
            int   m = cbase + j;
            if (v < dl[KNN-1]) {            // EXEC-predicated; wave re-converges after
                dl[KNN-1] = v;
                il[KNN-1] = m;
#pragma unroll
                for (int q = KNN-1; q >= 1; --q) {
                    float dq = dl[q];
                    if (dq < dl[q-1]) {
                        dl[q] = dl[q-1]; dl[q-1] = dq;
                        int tq = il[q]; il[q] = il[q-1]; il[q-1] = tq;
                    }
                }
            }
        }
        bv = bnext;
    }

    // ---- Merge lo/hi half-lists: smallest 16 of two sorted 16-lists is
    //      the multiset { min(A[i], B[15-i]) }.  Lo lane l pairs with lane l+16. ----
    float md[KNN];
    int   mi[KNN];
#pragma unroll
    for (int i = 0; i < KNN; ++i) {
        float od = __shfl(dl[KNN-1-i], lane | 16, 32);
        int   oi = __shfl(il[KNN-1-i], lane | 16, 32);
        bool take = od < dl[i];
        md[i] = take ? od : dl[i];
        mi[i] = take ? oi : il[i];
    }

    // ---- Final phase (lo lanes only): find nearest (center), min over 15 edges ----
    if (hi == 0) {
        float bestd = md[0];
        int   besti = mi[0];
        int   bestslot = 0;
#pragma unroll
        for (int i = 1; i < KNN; ++i) {
            if (md[i] < bestd) { bestd = md[i]; besti = mi[i]; bestslot = i; }
        }

        const float cx = -0.5f * sraw[4*besti+0];
        const float cy = -0.5f * sraw[4*besti+1];
        const float cz = -0.5f * sraw[4*besti+2];
        const float tx = px - cx, ty = py - cy, tz = pz - cz;

        float mn = 3.0e38f;
#pragma unroll
        for (int i = 0; i < KNN; ++i) {
            if (i != bestslot) {
                int ii = mi[i];
                float ex = -0.5f * sraw[4*ii+0] - cx;
                float ey = -0.5f * sraw[4*ii+1] - cy;
                float ez = -0.5f * sraw[4*ii+2] - cz;
                float el2 = ex*ex + ey*ey + ez*ez;
                float el  = sqrtf(el2);
                float vl  = (tx*ex + ty*ey + tz*ez) / el;
                float dd  = vl - 0.5f * el;
                mn = fminf(mn, dd * dd);
            }
        }
        out[(size_t)b * NPTS + row] = mn;
    }
}

extern "C" void kernel_launch(void* const* d_in, const int* in_sizes, int n_in,
                              void* d_out, int out_size, void* d_ws, size_t ws_size,
                              hipStream_t stream) {
    (void)in_sizes; (void)n_in; (void)d_ws; (void)ws_size; (void)out_size;
    const float* points  = (const float*)d_in[0];   // (2, 8192, 3) f32
    const float* spoints = (const float*)d_in[1];   // (2, 8192, 3) f32
    float* out = (float*)d_out;                     // (2, 8192) f32

    dim3 grid(NPTS / PTS_PER_BLOCK, 2);
    dim3 block(BLOCK_THREADS);
    size_t lds_bytes = (size_t)NPTS * 4 * sizeof(float);   // 128 KB (<= 320 KB/WG on CDNA5)
    voroloss_kernel<<<grid, block, lds_bytes, stream>>>(points, spoints, out);
}